// Loss_65996467471179
// MI455X (gfx1250) — compile-verified
//
#include <hip/hip_runtime.h>
#include <math.h>

#define NUM_C   65536
#define L_PTS   256
#define N_TOT   (NUM_C * L_PTS)

typedef __attribute__((ext_vector_type(2))) float v2f;
typedef __attribute__((ext_vector_type(8))) float v8f;

__device__ __forceinline__ float relu(float x) { return fmaxf(x, 0.0f); }

// ---------------------------------------------------------------------------
// Kernel 1: one wave32 per curve. Streams the five big arrays (336 MB total),
// does the per-curve segmented reductions + argmin, writes one partial/curve.
// Bandwidth-bound: ~14.5 us floor at 23.3 TB/s.
// ---------------------------------------------------------------------------
__global__ __launch_bounds__(256) void curve_kernel(
    const float* __restrict__ An, const float* __restrict__ Ac,
    const float* __restrict__ Aj, const float* __restrict__ Ap,
    const float* __restrict__ Ar, const float* __restrict__ Ci,
    const int*   __restrict__ maskL,
    float* __restrict__ ws_partial)
{
    const int lane = threadIdx.x & 31;
    const int wv   = threadIdx.x >> 5;
    const int c    = blockIdx.x * 8 + wv;     // curve index
    const int base = c * L_PTS;

    const float4* An4 = reinterpret_cast<const float4*>(An + base);
    const float4* Ac4 = reinterpret_cast<const float4*>(Ac + base);
    const float4* Aj4 = reinterpret_cast<const float4*>(Aj + base);
    const float4* Ap4 = reinterpret_cast<const float4*>(Ap + base);
    const float4* Ar4 = reinterpret_cast<const float4*>(Ar + base);

    float mse = 0.f, apsum = 0.f, lsaj = 0.f, lsac = 0.f;
    float amin = 3.4e38f;
    int   aidx = 0;

#pragma unroll
    for (int v = 0; v < 2; ++v) {
        const int vi = lane * 2 + v;          // lane owns points [lane*8, lane*8+8)
        float4 an = An4[vi], ac = Ac4[vi], aj = Aj4[vi], ap = Ap4[vi], ar = Ar4[vi];
        const float* pan = &an.x; const float* pac = &ac.x;
        const float* paj = &aj.x; const float* pap = &ap.x;
        const float* par = &ar.x;
#pragma unroll
        for (int t = 0; t < 4; ++t) {
            const int j = lane * 8 + v * 4 + t;
            float d = pan[t] - par[t];
            mse += d * d;
            apsum += relu(-pap[t]);
            float acj = pac[t] - paj[t];
            lsaj += relu(-acj);
            lsac += relu(acj);
            float a = fabsf(acj);
            if (a < amin) { amin = a; aidx = j; }   // in-lane order => first min
        }
    }

    // wave32 reductions (sums + argmin with first-index tie-break)
#pragma unroll
    for (int off = 16; off > 0; off >>= 1) {
        mse   += __shfl_xor(mse,   off, 32);
        apsum += __shfl_xor(apsum, off, 32);
        lsaj  += __shfl_xor(lsaj,  off, 32);
        lsac  += __shfl_xor(lsac,  off, 32);
        float oa = __shfl_xor(amin, off, 32);
        int   oi = __shfl_xor(aidx, off, 32);
        if (oa < amin || (oa == amin && oi < aidx)) { amin = oa; aidx = oi; }
    }

    if (lane == 0) {
        const int e = base + (L_PTS - 1);
        float aj_e = Aj[e], ac_e = Ac[e], ap_e = Ap[e], ci_e = Ci[e];
        int   m = maskL[c];
        float w = (m == 0) ? 1.0f : 0.0f;

        float aj_i = Aj[base + aidx];          // crossover gather (L1/L2 hit)
        float ap_i = Ap[base + aidx];

        float pen = 3.0f * relu(aj_i * 1.1f - ap_i);
        pen += w * (relu(8.0f - lsaj) + relu(8.0f - lsac));
        pen += relu(aj_e - ac_e);
        if (ci_e > 500.0f && m == 0) pen += 0.15f * relu(ap_e - aj_e);

        ws_partial[c] = mse * (10.0f / (float)N_TOT) + apsum + pen;
    }
}

// ---------------------------------------------------------------------------
// Kernel 2: cross-curve combine. Eight f32 sums over C=65536 elements are
// accumulated with V_WMMA_F32_16X16X4_F32 (A == ones => D += column-sums of B;
// 64 exact f32 adds per matrix op). Total = sum of D row 0 (VGPR0, lanes 0-15).
// ---------------------------------------------------------------------------
__global__ __launch_bounds__(256) void finalize_kernel(
    const float* __restrict__ ws_partial,
    const float* __restrict__ Vc,  const float* __restrict__ Jm,
    const float* __restrict__ Rd,  const int*   __restrict__ maskL,
    const float* __restrict__ dHaV, const float* __restrict__ dHaJ,
    const float* __restrict__ dHaT,
    const float* __restrict__ ToV,  const float* __restrict__ ToJ,
    const float* __restrict__ ToT,
    float* __restrict__ out)
{
    const int lane = threadIdx.x & 31;
    const int wv   = threadIdx.x >> 5;

    v8f acc[8];
#pragma unroll
    for (int q = 0; q < 8; ++q) acc[q] = (v8f){0.f,0.f,0.f,0.f,0.f,0.f,0.f,0.f};

    const v2f ones = {1.0f, 1.0f};
    const int per   = NUM_C / 8;               // 8192 elements per wave
    const int start = wv * per;
    const int stop  = start + per;

    for (int b = start; b < stop; b += 64) {   // uniform per-wave loop: EXEC all-1s
        float qq[8][2];
#pragma unroll
        for (int p = 0; p < 2; ++p) {
            const int i = b + p * 32 + lane;
            float ws = ws_partial[i];
            float w  = (maskL[i] == 0) ? 1.0f : 0.0f;
            float x  = Jm[i];
            float y  = Vc[i];
            qq[0][p] = ws;
            qq[1][p] = w;
            qq[2][p] = w * x;
            qq[3][p] = w * y;
            qq[4][p] = w * x * x;
            qq[5][p] = w * y * y;
            qq[6][p] = w * x * y;
            qq[7][p] = relu(-Rd[i]);
        }
#pragma unroll
        for (int q = 0; q < 8; ++q) {
            v2f bq = {qq[q][0], qq[q][1]};
            acc[q] = __builtin_amdgcn_wmma_f32_16x16x4_f32(
                false, ones, false, bq, (short)0, acc[q], false, false);
        }
    }

    // Extract: D[0][n] sits in VGPR0 (component 0) lanes 0..15; 16-lane xor-add.
    __shared__ float sums[8][8];
#pragma unroll
    for (int q = 0; q < 8; ++q) {
        float s = acc[q][0];
        s += __shfl_xor(s, 1, 32);
        s += __shfl_xor(s, 2, 32);
        s += __shfl_xor(s, 4, 32);
        s += __shfl_xor(s, 8, 32);
        if (lane == 0) sums[wv][q] = s;
    }
    __syncthreads();

    if (threadIdx.x == 0) {
        float T[8];
#pragma unroll
        for (int q = 0; q < 8; ++q) {
            float s = 0.f;
#pragma unroll
            for (int k = 0; k < 8; ++k) s += sums[k][q];
            T[q] = s;
        }
        float loss = T[0];                     // MSE*10 + relu(-Ap) + per-curve penalties
        float nw = T[1], Sx = T[2], Sy = T[3], Sxx = T[4], Syy = T[5], Sxy = T[6];

        float cov = Sxy - Sx * Sy / nw;
        float vxx = Sxx - Sx * Sx / nw;
        float vyy = Syy - Sy * Sy / nw;
        float corr = cov / (sqrtf(vxx) * sqrtf(vyy));
        if (!(corr == corr)) corr = 0.0f;      // NaN -> 0
        float cost = fminf(corr, 0.7f);
        loss += (0.7f - cost);
        loss += T[7];                          // sum relu(-Rd25)

#pragma unroll
        for (int i = 0; i < 4; ++i) {
            loss += 10.0f * relu(-dHaV[i]) + relu(-dHaJ[i]) + relu(-dHaT[i]);
            loss += relu(273.15f - ToV[i]) + relu(273.15f - ToJ[i]) + relu(273.15f - ToT[i]);
        }
        out[0] = loss;
    }
}

extern "C" void kernel_launch(void* const* d_in, const int* in_sizes, int n_in,
                              void* d_out, int out_size, void* d_ws, size_t ws_size,
                              hipStream_t stream) {
    const float* An   = (const float*)d_in[0];
    const float* Ac   = (const float*)d_in[1];
    const float* Aj   = (const float*)d_in[2];
    const float* Ap   = (const float*)d_in[3];
    const float* Ar   = (const float*)d_in[4];
    const float* Ci   = (const float*)d_in[5];
    const float* Vc   = (const float*)d_in[6];
    const float* Jm   = (const float*)d_in[7];
    const float* Rd   = (const float*)d_in[8];
    const float* dHaV = (const float*)d_in[9];
    const float* dHaJ = (const float*)d_in[10];
    const float* dHaT = (const float*)d_in[11];
    const float* ToV  = (const float*)d_in[12];
    const float* ToJ  = (const float*)d_in[13];
    const float* ToT  = (const float*)d_in[14];
    const int*   mask = (const int*)d_in[15];

    float* ws  = (float*)d_ws;      // NUM_C partials = 256 KB
    float* out = (float*)d_out;

    curve_kernel<<<NUM_C / 8, 256, 0, stream>>>(An, Ac, Aj, Ap, Ar, Ci, mask, ws);
    finalize_kernel<<<1, 256, 0, stream>>>(ws, Vc, Jm, Rd, mask,
                                           dHaV, dHaJ, dHaT, ToV, ToJ, ToT, out);
}